// ForwardModel_54116587930190
// MI455X (gfx1250) — compile-verified
//
#include <hip/hip_runtime.h>
#include <math.h>

typedef float v2f __attribute__((ext_vector_type(2)));
typedef float v8f __attribute__((ext_vector_type(8)));

#define NT 256

// ---- constants matching the reference ----
constexpr float HBAR_F   = 0.06465390984476556f;
constexpr float KB_F     = 8.617333262e-05f;
constexpr float T_F      = 10.0f;
constexpr float NCELLS_F = 27.0f;
constexpr float MASS_F   = 26.981538f;
constexpr float INV2KBT  = 1.0f / (2.0f * KB_F * T_F);          // multiplies sqrt(|lam|)
constexpr float MS_COEF  = HBAR_F / (4.0f * MASS_F * NCELLS_F);
constexpr float INV_MASS = 1.0f / MASS_F;
constexpr float DELTA_PREF = 1.0f / (3.14159265358979323846f * 1e-5f);
constexpr float INV_SIG2 = 1.0e10f;                              // 1/A_SIGMA^2
constexpr float CVAL_F   = 1.1151f;
__device__ __constant__ float AV[4] = {6.4202f, 1.9002f, 1.5936f, 1.9646f};
__device__ __constant__ float BV[4] = {3.0387f, 426.0f, 31.5472f, 85.0886f};

// ---- tiny complex helpers ----
struct cpx { float re, im; };
__device__ __forceinline__ cpx cmul(cpx a, cpx b){ return {a.re*b.re - a.im*b.im, a.re*b.im + a.im*b.re}; }
__device__ __forceinline__ cpx csub(cpx a, cpx b){ return {a.re-b.re, a.im-b.im}; }
__device__ __forceinline__ cpx cconj(cpx a){ return {a.re, -a.im}; }
__device__ __forceinline__ float cabs2(cpx a){ return a.re*a.re + a.im*a.im; }
__device__ __forceinline__ cpx cscale(cpx a, float s){ return {a.re*s, a.im*s}; }

__device__ __forceinline__ void crossc(const cpx u[3], const cpx w[3], cpx o[3]) {
  o[0] = csub(cmul(u[1], w[2]), cmul(u[2], w[1]));
  o[1] = csub(cmul(u[2], w[0]), cmul(u[0], w[2]));
  o[2] = csub(cmul(u[0], w[1]), cmul(u[1], w[0]));
}

// Eigenvector of 3x3 Hermitian for eigenvalue lam: cross product of rows of (H - lam I).
__device__ __forceinline__ void eigvec3(float a0, float a1, float a2,
                                        cpx d, cpx e, cpx f, float lam, cpx v[3]) {
  cpx r0[3] = {{a0 - lam, 0.f}, d, e};
  cpx r1[3] = {cconj(d), {a1 - lam, 0.f}, f};
  cpx r2[3] = {cconj(e), cconj(f), {a2 - lam, 0.f}};
  cpx c0[3], c1[3], c2[3];
  crossc(r0, r1, c0); crossc(r1, r2, c1); crossc(r2, r0, c2);
  float n0 = cabs2(c0[0]) + cabs2(c0[1]) + cabs2(c0[2]);
  float n1 = cabs2(c1[0]) + cabs2(c1[1]) + cabs2(c1[2]);
  float n2 = cabs2(c2[0]) + cabs2(c2[1]) + cabs2(c2[2]);
  cpx b0 = c0[0], b1 = c0[1], b2 = c0[2]; float nb = n0;
  if (n1 > nb) { b0 = c1[0]; b1 = c1[1]; b2 = c1[2]; nb = n1; }
  if (n2 > nb) { b0 = c2[0]; b1 = c2[1]; b2 = c2[2]; nb = n2; }
  float inv = rsqrtf(nb + 1e-30f);
  v[0] = cscale(b0, inv); v[1] = cscale(b1, inv); v[2] = cscale(b2, inv);
  // deterministic phase: make largest-|.| component real positive
  float m0 = cabs2(v[0]), m1 = cabs2(v[1]), m2 = cabs2(v[2]);
  cpx vm = v[0]; float mm = m0;
  if (m1 > mm) { vm = v[1]; mm = m1; }
  if (m2 > mm) { vm = v[2]; mm = m2; }
  cpx ph = cscale(cconj(vm), rsqrtf(mm + 1e-30f));
  v[0] = cmul(v[0], ph); v[1] = cmul(v[1], ph); v[2] = cmul(v[2], ph);
}

// ---- 32-lane f32 sum on the matrix pipe: V_WMMA_F32_16X16X4_F32 ----
// A(16x4): lanes 0-15 vgpr0 -> A[m][0], vgpr1 -> A[m][1]; lanes 16-31 -> A[m][2], A[m][3].
// With x in vgpr0, 0 in vgpr1: rowsum[m] = x_m + x_{m+16}. B = ones => D[m][n] = rowsum[m].
// Each lane sums its 8 D accumulators (covers 8 of 16 rows), then a half-swap shuffle
// combines the two halves -> full 32-lane sum in every lane. EXEC must be all ones.
__device__ __forceinline__ float wave_sum_wmma(float x) {
  v2f a; a[0] = x;    a[1] = 0.0f;
  v2f b; b[0] = 1.0f; b[1] = 1.0f;
  v8f c = {};
  v8f dm = __builtin_amdgcn_wmma_f32_16x16x4_f32(false, a, false, b, (short)0, c, false, false);
  float s = dm[0] + dm[1] + dm[2] + dm[3] + dm[4] + dm[5] + dm[6] + dm[7];
  s += __shfl_xor(s, 16, 32);
  return s;
}

// ===================== K1: eigensolve, c_indv, per-block Re(BZ) partials =====================
__global__ void __launch_bounds__(NT)
k1_eig(const float* __restrict__ qin, const float* __restrict__ wmat,
       const float* __restrict__ bmat, float* __restrict__ c_indv,
       float* __restrict__ rbzPart, int nk, int nblk) {
  int k = blockIdx.x * NT + threadIdx.x;
  bool valid = (k < nk);
  int kk = valid ? k : (nk - 1);           // clamp: keep all lanes active (WMMA needs EXEC=~0)
  const float* wp = wmat + (size_t)kk * 9;
  const float* bp = bmat + (size_t)kk * 9;
  // H = (A + A^H)/2, A = W + iB
  float a0 = wp[0], a1 = wp[4], a2 = wp[8];
  cpx d = {0.5f * (wp[1] + wp[3]), 0.5f * (bp[1] - bp[3])};
  cpx e = {0.5f * (wp[2] + wp[6]), 0.5f * (bp[2] - bp[6])};
  cpx f = {0.5f * (wp[5] + wp[7]), 0.5f * (bp[5] - bp[7])};
  // analytic eigenvalues (trigonometric cubic)
  float p1 = cabs2(d) + cabs2(e) + cabs2(f);
  float qm = (a0 + a1 + a2) * (1.0f / 3.0f);
  float g0 = a0 - qm, g1 = a1 - qm, g2 = a2 - qm;
  float p2 = g0 * g0 + g1 * g1 + g2 * g2 + 2.0f * p1;
  float p  = sqrtf(p2 * (1.0f / 6.0f));
  float ip = 1.0f / fmaxf(p, 1e-20f);
  float h0 = g0 * ip, h1 = g1 * ip, h2 = g2 * ip;
  cpx dp = cscale(d, ip), ep = cscale(e, ip), fp = cscale(f, ip);
  cpx t = cmul(dp, fp);
  float detB = h0 * h1 * h2 - h0 * cabs2(fp) - h1 * cabs2(ep) - h2 * cabs2(dp)
             + 2.0f * (t.re * ep.re + t.im * ep.im);
  float r = fminf(fmaxf(0.5f * detB, -1.0f), 1.0f);
  float phi = acosf(r) * (1.0f / 3.0f);
  float lamA = qm + 2.0f * p * cosf(phi);                        // largest
  float lamC = qm + 2.0f * p * cosf(phi + 2.0943951023931953f);  // smallest
  float lamB = 3.0f * qm - lamA - lamC;
  float lam[3] = {lamC, lamB, lamA};                             // ascending, like eigh
  float D[3];
#pragma unroll
  for (int j = 0; j < 3; ++j) {
    float s = sqrtf(fabsf(lam[j]));
    float den = s * tanhf(s * INV2KBT);          // D = HBAR / (sqrt|lam| * tanh(...))
    D[j] = HBAR_F / fmaxf(den, 1e-30f);
  }
  cpx V[3][3];                                   // V[mode][component]
  eigvec3(a0, a1, a2, d, e, f, lam[0], V[0]);
  eigvec3(a0, a1, a2, d, e, f, lam[1], V[1]);
  eigvec3(a0, a1, a2, d, e, f, lam[2], V[2]);
  // c_indv = sum_j D_j |u_j|^2, u = q0*v0 + q1*v1 + q2*v2  (BZ = V^H D V, q real)
  const float* qp = qin + (size_t)kk * 9;
  float q0 = qp[0], q1 = qp[1], q2 = qp[2];
  float cind = 0.0f;
#pragma unroll
  for (int j = 0; j < 3; ++j) {
    cpx u = {q0 * V[0][j].re + q1 * V[1][j].re + q2 * V[2][j].re,
             q0 * V[0][j].im + q1 * V[1][j].im + q2 * V[2][j].im};
    cind += D[j] * cabs2(u);
  }
  // Re(BZ)_{ab} = sum_j D_j (Re v_a[j] Re v_b[j] + Im v_a[j] Im v_b[j]) ; 6 unique entries
  const int pa[6] = {0, 0, 0, 1, 1, 2};
  const int pb[6] = {0, 1, 2, 1, 2, 2};
  float rbz[6];
#pragma unroll
  for (int pI = 0; pI < 6; ++pI) {
    int aI = pa[pI], bI = pb[pI];
    float s = 0.0f;
#pragma unroll
    for (int j = 0; j < 3; ++j)
      s += D[j] * (V[aI][j].re * V[bI][j].re + V[aI][j].im * V[bI][j].im);
    rbz[pI] = valid ? s : 0.0f;
  }
  if (valid) c_indv[k] = cind;
  // wave reduction on the matrix pipe, then cross-wave via LDS
  __shared__ float lds[NT / 32][6];
  float s6[6];
#pragma unroll
  for (int pI = 0; pI < 6; ++pI) s6[pI] = wave_sum_wmma(rbz[pI]);
  int lane = threadIdx.x & 31, wv = threadIdx.x >> 5;
  if (lane == 0)
    for (int pI = 0; pI < 6; ++pI) lds[wv][pI] = s6[pI];
  __syncthreads();
  if (threadIdx.x == 0) {
    for (int pI = 0; pI < 6; ++pI) {
      float acc = 0.0f;
      for (int w8 = 0; w8 < NT / 32; ++w8) acc += lds[w8][pI];
      rbzPart[(size_t)pI * nblk + blockIdx.x] = acc;
    }
  }
}

// ===================== K2/K4: deterministic tree reduction of `rows` arrays of length n ==========
__global__ void __launch_bounds__(NT)
reduce_cols(const float* __restrict__ in, float* __restrict__ out, int n, int rows) {
  __shared__ float red[NT];
  for (int rI = 0; rI < rows; ++rI) {
    float s = 0.0f;
    for (int i = threadIdx.x; i < n; i += NT) s += in[(size_t)rI * n + i];
    red[threadIdx.x] = s;
    __syncthreads();
    for (int st = NT / 2; st > 0; st >>= 1) {
      if (threadIdx.x < st) red[threadIdx.x] += red[threadIdx.x + st];
      __syncthreads();
    }
    if (threadIdx.x == 0) out[rI] = red[0];
    __syncthreads();
  }
}

// ===================== K3: Itot -> d_out, per-block S partials =====================
__global__ void __launch_bounds__(NT)
k3_itot(const float* __restrict__ qin, const float* __restrict__ c_indv,
        const float* __restrict__ sumBz, float* __restrict__ itot_out,
        float* __restrict__ SPart, int nk, int nblk) {
  int k = blockIdx.x * NT + threadIdx.x;
  bool valid = (k < nk);
  int kk = valid ? k : (nk - 1);
  const float* qp = qin + (size_t)kk * 9;
  float q0 = qp[0], q1 = qp[1], q2 = qp[2];
  float k0 = qp[3], k1 = qp[4], k2 = qp[5];
  float q2s = q0 * q0 + q1 * q1 + q2 * q2;
  float fs = CVAL_F;
#pragma unroll
  for (int i = 0; i < 4; ++i) fs += AV[i] * expf(-BV[i] * q2s);
  float ksq = k0 * k0 + k1 * k1 + k2 * k2;
  float delta = DELTA_PREF * expf(-ksq * INV_SIG2);
  float S00 = sumBz[0], S01 = sumBz[1], S02 = sumBz[2];
  float S11 = sumBz[3], S12 = sumBz[4], S22 = sumBz[5];
  float cs = S00 * q0 * q0 + S11 * q1 * q1 + S22 * q2 * q2
           + 2.0f * (S01 * q0 * q1 + S02 * q0 * q2 + S12 * q1 * q2);
  float Ms = MS_COEF * cs;
  float itot = fs * fs * expf(-2.0f * Ms) * (delta + c_indv[kk] * INV_MASS);
  if (valid) itot_out[k] = itot;
  float contrib = valid ? itot : 0.0f;
  __shared__ float lds[NT / 32];
  float s = wave_sum_wmma(contrib);
  int lane = threadIdx.x & 31, wv = threadIdx.x >> 5;
  if (lane == 0) lds[wv] = s;
  __syncthreads();
  if (threadIdx.x == 0) {
    float acc = 0.0f;
    for (int w8 = 0; w8 < NT / 32; ++w8) acc += lds[w8];
    SPart[blockIdx.x] = acc;
  }
}

// ===================== K5: normalize =====================
__global__ void __launch_bounds__(NT)
k5_norm(float* __restrict__ out, const float* __restrict__ Sval, int nk) {
  int k = blockIdx.x * NT + threadIdx.x;
  if (k < nk) out[k] = out[k] / Sval[0];
}

extern "C" void kernel_launch(void* const* d_in, const int* in_sizes, int n_in,
                              void* d_out, int out_size, void* d_ws, size_t ws_size,
                              hipStream_t stream) {
  const float* qin = (const float*)d_in[0];
  const float* wm  = (const float*)d_in[1];
  const float* bm  = (const float*)d_in[2];
  float* out = (float*)d_out;
  int nk   = in_sizes[0] / 9;
  int nblk = (nk + NT - 1) / NT;
  // workspace layout (floats): c_indv[nk] | rbzPart[6*nblk] | sumBz[6] | SPart[nblk] | Sval[1]
  float* ws      = (float*)d_ws;
  float* c_indv  = ws;
  float* rbzPart = c_indv + nk;
  float* sumBz   = rbzPart + (size_t)6 * nblk;
  float* SPart   = sumBz + 6;
  float* Sval    = SPart + nblk;

  k1_eig<<<nblk, NT, 0, stream>>>(qin, wm, bm, c_indv, rbzPart, nk, nblk);
  reduce_cols<<<1, NT, 0, stream>>>(rbzPart, sumBz, nblk, 6);
  k3_itot<<<nblk, NT, 0, stream>>>(qin, c_indv, sumBz, out, SPart, nk, nblk);
  reduce_cols<<<1, NT, 0, stream>>>(SPart, Sval, nblk, 1);
  k5_norm<<<nblk, NT, 0, stream>>>(out, Sval, nk);
}